// PreferenceLayer_10479720202246
// MI455X (gfx1250) — compile-verified
//
#include <hip/hip_runtime.h>

#define NUM_USERS 50000
#define NUM_ITEMS 50000
#define N_NODES   (NUM_USERS + NUM_ITEMS)
#define EMB_DIM   64
#define N_LAYERS  3

typedef __attribute__((ext_vector_type(2))) float v2f;
typedef __attribute__((ext_vector_type(8))) float v8f;

// ---------------------------------------------------------------------------
// init: cur = concat(user_emb, item_emb); acc = cur     (float4 vectorized)
// ---------------------------------------------------------------------------
__global__ void lgcn_init_kernel(const float* __restrict__ ue,
                                 const float* __restrict__ ie,
                                 float* __restrict__ cur,
                                 float* __restrict__ acc,
                                 int n4user, int n4total) {
  int i = blockIdx.x * blockDim.x + threadIdx.x;
  if (i >= n4total) return;
  float4 v = (i < n4user) ? ((const float4*)ue)[i]
                          : ((const float4*)ie)[i - n4user];
  ((float4*)cur)[i] = v;
  ((float4*)acc)[i] = v;
}

// ---------------------------------------------------------------------------
// zero: nxt = 0
// ---------------------------------------------------------------------------
__global__ void lgcn_zero_kernel(float* __restrict__ p, int n4) {
  int i = blockIdx.x * blockDim.x + threadIdx.x;
  if (i >= n4) return;
  ((float4*)p)[i] = make_float4(0.f, 0.f, 0.f, 0.f);
}

// ---------------------------------------------------------------------------
// scatter: one wave32 per edge; lane handles 2 dims (float2 gather, 2 atomics)
// 25.6MB table lives in L2 (192MB) -> gathers + atomics stay on-chip.
// ---------------------------------------------------------------------------
__global__ void lgcn_scatter_kernel(const float* __restrict__ cur,
                                    float* __restrict__ nxt,
                                    const float* __restrict__ ev,
                                    const int* __restrict__ es,
                                    const int* __restrict__ ed,
                                    int nnz) {
  long long idx = (long long)blockIdx.x * blockDim.x + threadIdx.x;
  int e    = (int)(idx >> 5);
  int lane = (int)(idx & 31);
  if (e >= nnz) return;
  int   src = es[e];
  int   dst = ed[e];
  float w   = ev[e];
  float2 m = ((const float2*)(cur + (long long)src * EMB_DIM))[lane];
  float* d = nxt + (long long)dst * EMB_DIM + lane * 2;
  unsafeAtomicAdd(d,     w * m.x);   // global_atomic_add_f32, no return
  unsafeAtomicAdd(d + 1, w * m.y);
}

// ---------------------------------------------------------------------------
// accumulate: acc += nxt
// ---------------------------------------------------------------------------
__global__ void lgcn_acc_kernel(float* __restrict__ acc,
                                const float* __restrict__ nxt, int n4) {
  int i = blockIdx.x * blockDim.x + threadIdx.x;
  if (i >= n4) return;
  float4 a = ((const float4*)acc)[i];
  float4 b = ((const float4*)nxt)[i];
  a.x += b.x; a.y += b.y; a.z += b.z; a.w += b.w;
  ((float4*)acc)[i] = a;
}

// ---------------------------------------------------------------------------
// final: gamma[b] = (1/16) * dot(acc[users[b]], acc[NUM_USERS+items[b]])
// One wave per 16-batch tile: D(16x16) = U(16x64) @ V^T(64x16) via 16 chained
// V_WMMA_F32_16X16X4_F32; diagonal of D is the 16 dot products.
// A layout (16x4 f32): lanes 0-15 row M=lane, K={0,1}; lanes 16-31 K={2,3}.
// B layout is the mirrored column form -> identical float2 fetch from V rows.
// ---------------------------------------------------------------------------
__global__ void __launch_bounds__(32)
lgcn_dot_wmma_kernel(const float* __restrict__ acc,
                     const int* __restrict__ users,
                     const int* __restrict__ items,
                     float* __restrict__ out) {
  int tile = blockIdx.x;
  int lane = threadIdx.x;        // 0..31, wave32
  int row  = lane & 15;
  int b    = tile * 16 + row;

  int u  = users[b];
  int it = items[b];
  const float* U = acc + (long long)u * EMB_DIM;
  const float* V = acc + (long long)(NUM_USERS + it) * EMB_DIM;

  int sub = (lane >> 4) << 1;    // 0 for lanes 0-15, 2 for lanes 16-31

  v8f c = {};
#pragma unroll
  for (int kk = 0; kk < 16; ++kk) {
    int koff = kk * 4 + sub;
    v2f a  = { U[koff], U[koff + 1] };
    v2f bb = { V[koff], V[koff + 1] };
    c = __builtin_amdgcn_wmma_f32_16x16x4_f32(
        /*neg_a=*/false, a, /*neg_b=*/false, bb,
        /*c_mod=*/(short)0, c, /*reuse_a=*/false, /*reuse_b=*/false);
  }

  // Extract diagonal D[m][m]:
  //   m in 0..7  -> VGPR m,   lane m
  //   m in 8..15 -> VGPR m-8, lane m+16  (i.e. lanes 24..31 hold c[lane-24])
  float g = 0.0f;
#pragma unroll
  for (int r = 0; r < 8; ++r) {
    if (lane == r)      g = c[r];
    if (lane == r + 24) g = c[r];
  }
  if (lane < 8) {
    out[tile * 16 + lane] = g * 0.0625f;               // (1/4)^2 layer mean
  } else if (lane >= 24) {
    out[tile * 16 + (lane - 16)] = g * 0.0625f;
  }
}

// ---------------------------------------------------------------------------
extern "C" void kernel_launch(void* const* d_in, const int* in_sizes, int n_in,
                              void* d_out, int out_size, void* d_ws, size_t ws_size,
                              hipStream_t stream) {
  const float* user_emb = (const float*)d_in[0];
  const float* item_emb = (const float*)d_in[1];
  const float* edge_val = (const float*)d_in[2];
  const int*   edge_src = (const int*)d_in[3];
  const int*   edge_dst = (const int*)d_in[4];
  const int*   users    = (const int*)d_in[5];
  const int*   items    = (const int*)d_in[6];
  float*       out      = (float*)d_out;

  const int nnz   = in_sizes[2];
  const int batch = in_sizes[5];

  const long long N64 = (long long)N_NODES * EMB_DIM;   // 6.4M floats
  float* bufA = (float*)d_ws;
  float* bufB = bufA + N64;
  float* acc  = bufB + N64;

  const int n4total = (int)(N64 / 4);                   // 1.6M float4
  const int n4user  = NUM_USERS * EMB_DIM / 4;

  const int TPB = 256;
  int gridN4 = (n4total + TPB - 1) / TPB;

  lgcn_init_kernel<<<gridN4, TPB, 0, stream>>>(user_emb, item_emb, bufA, acc,
                                               n4user, n4total);

  float* cur = bufA;
  float* nxt = bufB;
  long long scatterThreads = (long long)nnz * 32;
  int gridScatter = (int)((scatterThreads + TPB - 1) / TPB);

  for (int layer = 0; layer < N_LAYERS; ++layer) {
    lgcn_zero_kernel<<<gridN4, TPB, 0, stream>>>(nxt, n4total);
    lgcn_scatter_kernel<<<gridScatter, TPB, 0, stream>>>(cur, nxt, edge_val,
                                                         edge_src, edge_dst, nnz);
    lgcn_acc_kernel<<<gridN4, TPB, 0, stream>>>(acc, nxt, n4total);
    float* t = cur; cur = nxt; nxt = t;
  }

  lgcn_dot_wmma_kernel<<<batch / 16, 32, 0, stream>>>(acc, users, items, out);
}